// Linear_DS_agent_70248485093612
// MI455X (gfx1250) — compile-verified
//
#include <hip/hip_runtime.h>
#include <hip/hip_bf16.h>
#include <math.h>

typedef __attribute__((ext_vector_type(2))) float v2f;
typedef __attribute__((ext_vector_type(4))) float v4f;
typedef __attribute__((ext_vector_type(8))) float v8f;

#define HS   116
#define PADN 128
#define XST  17          // padded Xout row stride (17 coprime with 64 banks)
#define GPW  10          // groups (of 16 steps) per workgroup
#define NBLK 1250        // 1250 * 10 * 16 = 200000 steps
#define NSTEPS 200000

// ---------------------------------------------------------------------------
// prep: x0 = l2(relu(l1(target))); solve P y0 = x0 (Gauss-Jordan w/ pivoting);
//       emit padded P (128x128), d = diag(D), d^16, LP = lm_w @ P.
// ---------------------------------------------------------------------------
__global__ __launch_bounds__(256) void prep_kernel(
    const float* __restrict__ target, const float* __restrict__ D,
    const float* __restrict__ P,
    const float* __restrict__ l1w, const float* __restrict__ l1b,
    const float* __restrict__ l2w, const float* __restrict__ l2b,
    const float* __restrict__ lmw,
    float* __restrict__ Ppad, float* __restrict__ y0,
    float* __restrict__ dvec, float* __restrict__ d16v,
    float* __restrict__ LP)
{
  __shared__ float aug[HS * 120];   // [P | x0], stride 120 -> ~56 KB
  __shared__ float h[HS];
  __shared__ float fcol[HS];
  __shared__ int   pivr;
  __shared__ float pivv;
  const int tid = threadIdx.x;

  // h = relu(l1w @ target + l1b)
  if (tid < HS) {
    float v = l1w[tid * 2 + 0] * target[0] + l1w[tid * 2 + 1] * target[1] + l1b[tid];
    h[tid] = v > 0.0f ? v : 0.0f;
  }
  __syncthreads();

  // aug[:, 0:116] = P ; aug[:, 116] = x0 = l2w @ h + l2b
  for (int idx = tid; idx < HS * HS; idx += 256) {
    int i = idx / HS, j = idx - i * HS;
    aug[i * 120 + j] = P[idx];
  }
  if (tid < HS) {
    float s = l2b[tid];
    for (int k = 0; k < HS; ++k) s += l2w[tid * HS + k] * h[k];
    aug[tid * 120 + HS] = s;
  }
  __syncthreads();

  // Gauss-Jordan with partial pivoting (117 active columns)
  for (int k = 0; k < HS; ++k) {
    if (tid == 0) {
      int p = k; float mx = fabsf(aug[k * 120 + k]);
      for (int r = k + 1; r < HS; ++r) {
        float v = fabsf(aug[r * 120 + k]);
        if (v > mx) { mx = v; p = r; }
      }
      pivr = p;
    }
    __syncthreads();
    const int p = pivr;
    if (p != k) {
      for (int j = tid; j < HS + 1; j += 256) {
        float t = aug[k * 120 + j];
        aug[k * 120 + j] = aug[p * 120 + j];
        aug[p * 120 + j] = t;
      }
    }
    __syncthreads();
    if (tid == 0) pivv = aug[k * 120 + k];
    __syncthreads();
    const float inv = 1.0f / pivv;
    for (int j = tid; j < HS + 1; j += 256) aug[k * 120 + j] *= inv;
    if (tid < HS) fcol[tid] = (tid == k) ? 0.0f : aug[tid * 120 + k];
    __syncthreads();
    for (int idx = tid; idx < HS * (HS + 1); idx += 256) {
      int i = idx / (HS + 1), j = idx - i * (HS + 1);
      if (i != k) aug[i * 120 + j] -= fcol[i] * aug[k * 120 + j];
    }
    __syncthreads();
  }

  // outputs
  if (tid < HS) y0[tid] = aug[tid * 120 + HS];
  for (int idx = tid; idx < PADN * PADN; idx += 256) {
    int i = idx >> 7, j = idx & 127;
    Ppad[idx] = (i < HS && j < HS) ? P[i * HS + j] : 0.0f;
  }
  if (tid < HS) {
    float dk = D[tid * HS + tid];
    dvec[tid] = dk;
    float x = dk * dk;  // d^2
    x *= x;             // d^4
    x *= x;             // d^8
    x *= x;             // d^16
    d16v[tid] = x;
  }
  for (int idx = tid; idx < 2 * HS; idx += 256) {
    int c = idx / HS, j = idx - c * HS;
    float s = 0.0f;
    for (int i2 = 0; i2 < HS; ++i2) s += lmw[c * HS + i2] * P[i2 * HS + j];
    LP[idx] = s;
  }
}

// ---------------------------------------------------------------------------
// rollout: hidden[:, t0..t0+15] = Ppad @ Wt, Wt advanced by d^16 per group.
// 8 waves; wave w owns row-tile w of Ppad, A-fragments resident in VGPRs.
// Output drained with 128-bit non-temporal stores.
// ---------------------------------------------------------------------------
__global__ __launch_bounds__(256) void rollout_kernel(
    const float* __restrict__ Ppad, const float* __restrict__ y0v,
    const float* __restrict__ dvec, const float* __restrict__ d16v,
    const float* __restrict__ LPg,  const float* __restrict__ lmb,
    float* __restrict__ out_act, float* __restrict__ out_hid)
{
  __shared__ float Wt[PADN * 16];     // B matrix (K=128 x N=16)
  __shared__ float Xout[PADN * XST];  // D tiles (M=128 x N=16), padded rows
  __shared__ float LPs[2 * HS];
  __shared__ float d16s[HS];

  const int tid  = threadIdx.x;
  const int wave = tid >> 5;
  const int lane = tid & 31;
  const int m    = lane & 15;           // row within tile (A) / col (B,D)
  const int kb2  = (lane >> 4) * 2;     // K sub-offset for A/B fragments
  const long t00 = (long)blockIdx.x * (GPW * 16);

  for (int i = tid; i < 2 * HS; i += 256) LPs[i] = LPg[i];
  if (tid < HS) d16s[tid] = d16v[tid];

  // init Wt columns: w[k][j] = y0[k] * d[k]^(t00 + j); zero-pad rows 116..127
  if (tid < PADN) {
    if (tid < HS) {
      float dk = dvec[tid];
      float p = y0v[tid] * powf(dk, (float)t00);
      for (int j = 0; j < 16; ++j) { Wt[tid * 16 + j] = p; p *= dk; }
    } else {
      for (int j = 0; j < 16; ++j) Wt[tid * 16 + j] = 0.0f;
    }
  }

  // resident A fragments: rows wave*16+m of Ppad
  v2f af[32];
#pragma unroll
  for (int kb = 0; kb < 32; ++kb) {
    int row = wave * 16 + m;
    int k = kb * 4 + kb2;
    af[kb].x = Ppad[row * PADN + k];
    af[kb].y = Ppad[row * PADN + k + 1];
  }
  __syncthreads();

  for (int g = 0; g < GPW; ++g) {
    const int t0 = (int)t00 + g * 16;

    // GEMM tile: acc = Ppad[tile wave] x Wt  (32 chained 16x16x4 f32 WMMAs)
    v8f acc = {};
#pragma unroll
    for (int kb = 0; kb < 32; ++kb) {
      int k = kb * 4 + kb2;
      v2f bf;
      bf.x = Wt[k * 16 + m];
      bf.y = Wt[(k + 1) * 16 + m];
      acc = __builtin_amdgcn_wmma_f32_16x16x4_f32(
          false, af[kb], false, bf, (short)0, acc, false, false);
    }
#pragma unroll
    for (int v = 0; v < 8; ++v)
      Xout[(wave * 16 + v + (lane >> 4) * 8) * XST + m] = acc[v];

    // actions: 32 (step j, channel c) pairs, 8 lanes each, shfl reduce
    {
      int pair = tid >> 3, sub = tid & 7;
      int c = pair & 1, j = pair >> 1;
      float s = 0.0f;
      for (int k = sub; k < HS; k += 8) s += LPs[c * HS + k] * Wt[k * 16 + j];
      s += __shfl_xor(s, 1);
      s += __shfl_xor(s, 2);
      s += __shfl_xor(s, 4);
      if (sub == 0)
        __builtin_nontemporal_store(tanhf(s + lmb[c]),
                                    out_act + (t0 + j) * 2 + c);
    }
    __syncthreads();

    // stream hidden states with 128-bit NT stores; advance Wt by d^16
    for (int idx = tid; idx < (HS / 4) * 16; idx += 256) {  // 29*16 quads
      int j = idx / 29, q = idx - j * 29;
      int base = (q * 4) * XST + j;
      v4f val;
      val.x = Xout[base];
      val.y = Xout[base + XST];
      val.z = Xout[base + 2 * XST];
      val.w = Xout[base + 3 * XST];
      __builtin_nontemporal_store(
          val, (v4f*)(out_hid + (long)(t0 + j) * HS + q * 4));
    }
    for (int idx = tid; idx < HS * 16; idx += 256) Wt[idx] *= d16s[idx >> 4];
    __syncthreads();
  }
}

// ---------------------------------------------------------------------------
extern "C" void kernel_launch(void* const* d_in, const int* in_sizes, int n_in,
                              void* d_out, int out_size, void* d_ws, size_t ws_size,
                              hipStream_t stream) {
  const float* target = (const float*)d_in[0];
  const float* D      = (const float*)d_in[1];
  const float* P      = (const float*)d_in[2];
  const float* l1w    = (const float*)d_in[3];
  const float* l1b    = (const float*)d_in[4];
  const float* l2w    = (const float*)d_in[5];
  const float* l2b    = (const float*)d_in[6];
  const float* lmw    = (const float*)d_in[7];
  const float* lmb    = (const float*)d_in[8];

  float* out     = (float*)d_out;
  float* out_act = out;                 // 400000 floats
  float* out_hid = out + 2 * NSTEPS;    // 200000 * 116 floats

  float* w    = (float*)d_ws;
  float* Ppad = w;                      // 16384
  float* y0   = w + 16384;              // 128
  float* dv   = w + 16512;              // 128
  float* d16v = w + 16640;              // 128
  float* LP   = w + 16768;              // 256

  prep_kernel<<<1, 256, 0, stream>>>(target, D, P, l1w, l1b, l2w, l2b, lmw,
                                     Ppad, y0, dv, d16v, LP);
  rollout_kernel<<<NBLK, 256, 0, stream>>>(Ppad, y0, dv, d16v, LP, lmb,
                                           out_act, out_hid);
}